// ConstraintOptimizer_74294344286523
// MI455X (gfx1250) — compile-verified
//
#include <hip/hip_runtime.h>
#include <hip/hip_bf16.h>

#define BIGF 1e30f

// ---------------- CDNA5 async global->LDS staging ----------------
#if defined(__has_builtin)
#  if __has_builtin(__builtin_amdgcn_global_load_async_to_lds_b128)
#    define HAS_ASYNC_LDS 1
#  endif
#endif
#ifndef HAS_ASYNC_LDS
#  define HAS_ASYNC_LDS 0
#endif

#if HAS_ASYNC_LDS
typedef int v4i __attribute__((vector_size(4 * sizeof(int))));
typedef __attribute__((address_space(1))) v4i* global_v4i_ptr;
typedef __attribute__((address_space(3))) v4i* lds_v4i_ptr;

__device__ __forceinline__ void async_copy16(const void* g, void* l) {
  // GLOBAL_LOAD_ASYNC_TO_LDS_B128: per-lane 16B global -> LDS, tracked by ASYNCcnt
  __builtin_amdgcn_global_load_async_to_lds_b128(
      (global_v4i_ptr)(unsigned long long)g,
      (lds_v4i_ptr)(unsigned)(unsigned long long)l,
      0, 0);
}
__device__ __forceinline__ void async_wait0() {
#  if __has_builtin(__builtin_amdgcn_s_wait_asynccnt)
  __builtin_amdgcn_s_wait_asynccnt(0);
#  else
  asm volatile("s_wait_asynccnt 0x0" ::: "memory");
#  endif
}
#endif

// Stage one branch row: 768 floats of points + 256 ints of mask into LDS.
__device__ __forceinline__ void stage_branch_rows(const float* __restrict__ rp_row,
                                                  const int* __restrict__ rm_row,
                                                  float* s_pts, int* s_msk, int tid) {
#if HAS_ASYNC_LDS
  if (tid < 192) async_copy16(rp_row + tid * 4, s_pts + tid * 4);
  if (tid < 64)  async_copy16(rm_row + tid * 4, s_msk + tid * 4);
  async_wait0();
#else
  if (tid < 192)
    reinterpret_cast<float4*>(s_pts)[tid] = reinterpret_cast<const float4*>(rp_row)[tid];
  if (tid < 64)
    reinterpret_cast<int4*>(s_msk)[tid] = reinterpret_cast<const int4*>(rm_row)[tid];
#endif
  __syncthreads();
}

// Build per-branch segment tables in LDS:
//   s_sl[255] segment lengths (0 if masked), s_sm[255] valid flags,
//   s_cum[256] exact left-to-right prefix (matches jnp.cumsum order),
//   s_tcum[128] trajectory cumulative arc length, *s_mvj, *s_has.
__device__ void build_branch(const float* __restrict__ rp_row,
                             const int* __restrict__ rm_row,
                             const float* __restrict__ pos_row,
                             bool fw, int tid,
                             float* s_pts, int* s_msk, float* s_sl, int* s_sm,
                             float* s_cum, float* s_tcum, int* s_mvj, int* s_has) {
  stage_branch_rows(rp_row, rm_row, s_pts, s_msk, tid);

  if (tid < 255) {
    int i0 = fw ? tid : 255 - tid;
    int i1 = fw ? tid + 1 : 254 - tid;
    int m = (s_msk[i0] != 0) && (s_msk[i1] != 0);
    float svx = s_pts[3 * i1 + 0] - s_pts[3 * i0 + 0];
    float svy = s_pts[3 * i1 + 1] - s_pts[3 * i0 + 1];
    float svz = s_pts[3 * i1 + 2] - s_pts[3 * i0 + 2];
    s_sm[tid] = m;
    s_sl[tid] = m ? sqrtf(svx * svx + svy * svy + svz * svz) : 0.0f;
  }
  __syncthreads();

  if (tid == 0) {  // exact sequential scan (wave 0) — matches cumsum associativity
    float c = 0.0f;
    s_cum[0] = 0.0f;
    int mv = 0, has = 0;
    for (int k = 0; k < 255; ++k) {
      c += s_sl[k];
      s_cum[k + 1] = c;
      if (s_sm[k]) { mv = k; has = 1; }
    }
    *s_mvj = mv;
    *s_has = has;
  }
  if (tid == 32) {  // trajectory cumulative arc length (wave 1, runs concurrently)
    float c = 0.0f;
    s_tcum[0] = 0.0f;
    for (int t = 0; t < 127; ++t) {
      float dx = pos_row[3 * (t + 1) + 0] - pos_row[3 * t + 0];
      float dy = pos_row[3 * (t + 1) + 1] - pos_row[3 * t + 1];
      float dz = pos_row[3 * (t + 1) + 2] - pos_row[3 * t + 2];
      c += sqrtf(dx * dx + dy * dy + dz * dz);
      s_tcum[t + 1] = c;
    }
  }
  __syncthreads();
}

// Block argmin with first-index tie-break (matches jnp.argmin) over 256 slots.
__device__ __forceinline__ void argmin_reduce256(float* v, int* ix, int tid) {
  __syncthreads();
  for (int s = 128; s; s >>= 1) {
    if (tid < s) {
      float vo = v[tid + s]; int io = ix[tid + s];
      float vm = v[tid];     int im = ix[tid];
      if (vo < vm || (vo == vm && io < im)) { v[tid] = vo; ix[tid] = io; }
    }
    __syncthreads();
  }
}

// Map arc-length target -> projected point. j = last index with cum[j] <= target
// (== sum(target >= cum)-1 for the nondecreasing scan), clamped to mvj.
__device__ __forceinline__ void project_point(float target, bool fw, int mvj,
                                              const float* s_cum, const float* s_sl,
                                              const float* s_pts,
                                              float& px, float& py, float& pz) {
  int j = 0;
#pragma unroll
  for (int sft = 128; sft; sft >>= 1) {
    int c = j + sft;
    if (c < 256 && s_cum[c] <= target) j = c;
  }
  if (j > mvj) j = mvj;
  float tl = (target - s_cum[j]) / fmaxf(s_sl[j], 1e-9f);
  tl = fminf(fmaxf(tl, 0.0f), 1.0f);
  int i0 = fw ? j : 255 - j;
  int i1 = fw ? j + 1 : 254 - j;
  float ax = s_pts[3 * i0 + 0], ay = s_pts[3 * i0 + 1], az = s_pts[3 * i0 + 2];
  px = ax + tl * (s_pts[3 * i1 + 0] - ax);
  py = ay + tl * (s_pts[3 * i1 + 1] - ay);
  pz = az + tl * (s_pts[3 * i1 + 2] - az);
}

// ---------------- Kernel A: per-(sample, branch2) cost ----------------
__global__ __launch_bounds__(256) void branch_cost_kernel(
    const float* __restrict__ pos, const float* __restrict__ rp,
    const int* __restrict__ rm,
    float* __restrict__ ws_cost, float* __restrict__ ws_entry,
    int* __restrict__ ws_has) {
  __shared__ alignas(16) float s_pts[768];
  __shared__ alignas(16) int   s_msk[256];
  __shared__ float s_sl[256];
  __shared__ int   s_sm[256];
  __shared__ float s_cum[256];
  __shared__ float s_tcum[128];
  __shared__ float s_rv[256];
  __shared__ int   s_ri[256];
  __shared__ int   s_mvj, s_has;
  __shared__ float s_entry;

  const int tid = threadIdx.x;
  const int blk = blockIdx.x;          // n*32 + b2
  const int n   = blk >> 5;
  const int b2  = blk & 31;
  const bool fw = (b2 < 16);
  const int b   = b2 & 15;

  const float* rp_row  = rp  + (size_t)(n * 16 + b) * 768;
  const int*   rm_row  = rm  + (size_t)(n * 16 + b) * 256;
  const float* pos_row = pos + (size_t)n * 384;
  __builtin_prefetch(pos_row, 0, 3);   // global_prefetch_b8

  build_branch(rp_row, rm_row, pos_row, fw, tid,
               s_pts, s_msk, s_sl, s_sm, s_cum, s_tcum, &s_mvj, &s_has);

  // --- entry point: closest segment to p0 (t0/d2 for all segs, mask to BIG) ---
  const float p0x = pos_row[0], p0y = pos_row[1], p0z = pos_row[2];
  float d2 = BIGF;
  if (tid < 255) {
    int i0 = fw ? tid : 255 - tid;
    int i1 = fw ? tid + 1 : 254 - tid;
    float ax = s_pts[3 * i0 + 0], ay = s_pts[3 * i0 + 1], az = s_pts[3 * i0 + 2];
    float svx = s_pts[3 * i1 + 0] - ax, svy = s_pts[3 * i1 + 1] - ay, svz = s_pts[3 * i1 + 2] - az;
    float svdot = fmaxf(svx * svx + svy * svy + svz * svz, 1e-12f);
    float t0 = ((p0x - ax) * svx + (p0y - ay) * svy + (p0z - az) * svz) / svdot;
    t0 = fminf(fmaxf(t0, 0.0f), 1.0f);
    float dx = p0x - (ax + t0 * svx);
    float dy = p0y - (ay + t0 * svy);
    float dz = p0z - (az + t0 * svz);
    float dd = dx * dx + dy * dy + dz * dz;
    d2 = s_sm[tid] ? dd : BIGF;
  }
  s_rv[tid] = d2;
  s_ri[tid] = tid;
  argmin_reduce256(s_rv, s_ri, tid);
  if (tid == 0) {
    int s0 = s_ri[0];
    int i0 = fw ? s0 : 255 - s0;
    int i1 = fw ? s0 + 1 : 254 - s0;
    float ax = s_pts[3 * i0 + 0], ay = s_pts[3 * i0 + 1], az = s_pts[3 * i0 + 2];
    float svx = s_pts[3 * i1 + 0] - ax, svy = s_pts[3 * i1 + 1] - ay, svz = s_pts[3 * i1 + 2] - az;
    float svdot = fmaxf(svx * svx + svy * svy + svz * svz, 1e-12f);
    float t0 = ((p0x - ax) * svx + (p0y - ay) * svy + (p0z - az) * svz) / svdot;
    t0 = fminf(fmaxf(t0, 0.0f), 1.0f);
    s_entry = s_cum[s0] + t0 * s_sl[s0];
  }
  __syncthreads();

  const float entry = s_entry;
  const float total = s_cum[255];
  const int   mvj   = s_mvj;

  // --- per-timestep projection + distance ---
  float dist = 0.0f;
  if (tid < 128) {
    float target = fmaxf(fminf(entry + s_tcum[tid], total), 0.0f);
    float px, py, pz;
    project_point(target, fw, mvj, s_cum, s_sl, s_pts, px, py, pz);
    float dx = pos_row[3 * tid + 0] - px;
    float dy = pos_row[3 * tid + 1] - py;
    float dz = pos_row[3 * tid + 2] - pz;
    dist = sqrtf(dx * dx + dy * dy + dz * dz);
  }
  __syncthreads();
  s_rv[tid] = (tid < 128) ? dist : 0.0f;
  __syncthreads();
  for (int s = 64; s; s >>= 1) {
    if (tid < s) s_rv[tid] += s_rv[tid + s];
    __syncthreads();
  }
  if (tid == 0) {
    ws_cost[blk]  = s_has ? s_rv[0] : BIGF;
    ws_entry[blk] = entry;
    ws_has[blk]   = s_has;
  }
}

// ---------------- Kernel B: pick best branch, emit projected trajectory ----------------
__global__ __launch_bounds__(256) void select_project_kernel(
    const float* __restrict__ pos, const float* __restrict__ rp,
    const int* __restrict__ rm,
    const float* __restrict__ ws_cost, const float* __restrict__ ws_entry,
    const int* __restrict__ ws_has, float* __restrict__ out) {
  __shared__ alignas(16) float s_pts[768];
  __shared__ alignas(16) int   s_msk[256];
  __shared__ float s_sl[256];
  __shared__ int   s_sm[256];
  __shared__ float s_cum[256];
  __shared__ float s_tcum[128];
  __shared__ float s_rv[256];
  __shared__ int   s_ri[256];
  __shared__ int   s_mvj, s_has, s_best, s_any;

  const int tid = threadIdx.x;
  const int n   = blockIdx.x;

  s_rv[tid] = (tid < 32) ? ws_cost[n * 32 + tid] : BIGF;
  s_ri[tid] = tid;
  argmin_reduce256(s_rv, s_ri, tid);
  if (tid == 0) {
    s_best = s_ri[0];
    int any = 0;
    for (int k = 0; k < 32; ++k) any |= ws_has[n * 32 + k];
    s_any = any;
  }
  __syncthreads();

  const int  best = s_best;
  const int  any  = s_any;
  const bool fw   = (best < 16);
  const int  b    = best & 15;

  const float* rp_row  = rp  + (size_t)(n * 16 + b) * 768;
  const int*   rm_row  = rm  + (size_t)(n * 16 + b) * 256;
  const float* pos_row = pos + (size_t)n * 384;

  build_branch(rp_row, rm_row, pos_row, fw, tid,
               s_pts, s_msk, s_sl, s_sm, s_cum, s_tcum, &s_mvj, &s_has);

  const float entry = ws_entry[n * 32 + best];
  const float total = s_cum[255];
  const int   mvj   = s_mvj;

  if (tid < 128) {
    float ox, oy, oz;
    if (any) {
      float target = fmaxf(fminf(entry + s_tcum[tid], total), 0.0f);
      project_point(target, fw, mvj, s_cum, s_sl, s_pts, ox, oy, oz);
    } else {
      ox = pos_row[3 * tid + 0];
      oy = pos_row[3 * tid + 1];
      oz = pos_row[3 * tid + 2];
    }
    size_t o = ((size_t)n * 128 + tid) * 3;
    out[o + 0] = ox;
    out[o + 1] = oy;
    out[o + 2] = oz;
  }
}

extern "C" void kernel_launch(void* const* d_in, const int* in_sizes, int n_in,
                              void* d_out, int out_size, void* d_ws, size_t ws_size,
                              hipStream_t stream) {
  const float* pos = (const float*)d_in[0];   // (64,128,3) f32
  const float* rp  = (const float*)d_in[1];   // (64,16,256,3) f32
  const int*   rm  = (const int*)d_in[2];     // (64,16,256) bool -> int32

  float* wsf      = (float*)d_ws;
  float* ws_cost  = wsf;            // 2048 floats
  float* ws_entry = wsf + 2048;     // 2048 floats
  int*   ws_has   = (int*)(wsf + 4096);  // 2048 ints (24 KB total)

  branch_cost_kernel<<<64 * 32, 256, 0, stream>>>(pos, rp, rm, ws_cost, ws_entry, ws_has);
  select_project_kernel<<<64, 256, 0, stream>>>(pos, rp, rm, ws_cost, ws_entry, ws_has,
                                                (float*)d_out);
  (void)in_sizes; (void)n_in; (void)out_size; (void)ws_size;
}